// CHOMP_15006615733288
// MI455X (gfx1250) — compile-verified
//
#include <hip/hip_runtime.h>

typedef __attribute__((ext_vector_type(2))) float v2f;
typedef __attribute__((ext_vector_type(8))) float v8f;

namespace {
constexpr int   kT    = 50;     // trajectory points
constexpr int   kJ    = 7;      // joints
constexpr int   kIter = 20;     // optimization iterations
constexpr float kLR   = 0.1f;
constexpr float kWObs = 10.0f;

// workspace layout (in floats)
constexpr int WS_TRAJ = 0;      // 350 floats: current trajectory (50x7)
constexpr int WS_S1   = 512;    // 64 floats:  per-row sum of w (rows padded to 64)
constexpr int WS_SO   = 576;    // 192 floats: per-row sum of w*o (x,y,z)
}

// ---------------------------------------------------------------------------
// Init: traj0[t][j] = start[j] + (t/(T-1)) * (goal[j]-start[j]); zero accums.
// ---------------------------------------------------------------------------
__global__ __launch_bounds__(512) void chomp_init(const float* __restrict__ start,
                                                  const float* __restrict__ goal,
                                                  float* __restrict__ ws) {
    int i = threadIdx.x;
    if (i < kT * kJ) {
        int t = i / kJ, j = i % kJ;
        float a = (float)t / (float)(kT - 1);
        ws[WS_TRAJ + i] = a * (goal[j] - start[j]) + start[j];
    }
    if (i < 256) ws[WS_S1 + i] = 0.0f;  // covers S1 (64) + So (192)
}

// ---------------------------------------------------------------------------
// Obstacle-field gradient partial sums via V_WMMA_F32_16X16X4_F32.
//   A[row]  = [-2px, -2py, -2pz, 1]   (16x4 tile, grid.y picks row tile)
//   B[:,c]  = [ox, oy, oz, |o|^2]     (4x16 tile of obstacles)
//   C[row]  = |p_row|^2 broadcast
//   => D = |p|^2 + |o|^2 - 2 p.o  (the 16x16 d2 tile, one WMMA per tile)
// Weight is branch-free:  w = max(rsqrt(d2) - 1, 0)  ==  (1-d)/d if d<1 else 0
// Two column-tiles (32 obstacles) per loop trip -> 2 WMMAs back-to-back.
// ---------------------------------------------------------------------------
__global__ __launch_bounds__(256) void chomp_grad(const float* __restrict__ obstacles,
                                                  const float* __restrict__ traj,
                                                  float* __restrict__ S1,
                                                  float* __restrict__ So,
                                                  int nObs, int nChunks) {
    const int lane  = threadIdx.x & 31;
    const int wid   = threadIdx.x >> 5;
    const int mtile = blockIdx.y;          // which block of 16 trajectory rows
    const int hi    = lane >> 4;           // half-wave: K={0,1} vs K={2,3}
    const int ln    = lane & 15;           // column (and A-row) within tile

    // A operand: row = mtile*16 + ln ; lanes<16 hold K0,K1; lanes>=16 hold K2,K3
    int arow = mtile * 16 + ln;
    v2f a;
    if (arow < kT) {
        if (hi == 0) {
            a.x = -2.0f * traj[arow * kJ + 0];
            a.y = -2.0f * traj[arow * kJ + 1];
        } else {
            a.x = -2.0f * traj[arow * kJ + 2];
            a.y = 1.0f;
        }
    } else {
        a.x = 0.0f;
        a.y = hi ? 1.0f : 0.0f;   // padded rows: d2 = |o|^2 (discarded later)
    }

    // C operand: |p|^2 for row = mtile*16 + r + 8*hi (matches 16x16 f32 C/D layout)
    v8f cmat;
#pragma unroll
    for (int r = 0; r < 8; ++r) {
        int row = mtile * 16 + r + 8 * hi;
        float s = 0.0f;
        if (row < kT) {
            float px = traj[row * kJ + 0];
            float py = traj[row * kJ + 1];
            float pz = traj[row * kJ + 2];
            s = px * px + py * py + pz * pz;
        }
        cmat[r] = s;
    }

    float accS1[8], accX[8], accY[8], accZ[8];
#pragma unroll
    for (int r = 0; r < 8; ++r) { accS1[r] = 0.f; accX[r] = 0.f; accY[r] = 0.f; accZ[r] = 0.f; }

    const int wavesTotal = gridDim.x * (blockDim.x >> 5);
    const int wglobal    = blockIdx.x * (blockDim.x >> 5) + wid;

    for (int c = wglobal; c < nChunks; c += wavesTotal) {
        const int col0 = c * 32 + ln;
        const int col1 = col0 + 16;
        // Branch-free bounds handling: clamp the address, poison |o|^2 so w==0.
        const int cc0 = min(col0, nObs - 1);
        const int cc1 = min(col1, nObs - 1);
        float ox0 = obstacles[cc0 * 3 + 0];
        float oy0 = obstacles[cc0 * 3 + 1];
        float oz0 = obstacles[cc0 * 3 + 2];
        float ox1 = obstacles[cc1 * 3 + 0];
        float oy1 = obstacles[cc1 * 3 + 1];
        float oz1 = obstacles[cc1 * 3 + 2];
        float os0 = (col0 < nObs) ? (ox0 * ox0 + oy0 * oy0 + oz0 * oz0) : 1e30f;
        float os1 = (col1 < nObs) ? (ox1 * ox1 + oy1 * oy1 + oz1 * oz1) : 1e30f;

        v2f b0; b0.x = hi ? oz0 : ox0; b0.y = hi ? os0 : oy0;
        v2f b1; b1.x = hi ? oz1 : ox1; b1.y = hi ? os1 : oy1;

        v8f d0 = __builtin_amdgcn_wmma_f32_16x16x4_f32(
            false, a, false, b0, (short)0, cmat, false, false);
        v8f d1 = __builtin_amdgcn_wmma_f32_16x16x4_f32(
            false, a, false, b1, (short)0, cmat, false, false);

#pragma unroll
        for (int r = 0; r < 8; ++r) {
            {   // tile 0
                float q  = fmaxf(d0[r], 1e-12f);
                float rs = __builtin_amdgcn_rsqf(q);
                rs = rs * (1.5f - 0.5f * q * rs * rs);     // one NR step -> ~fp32 exact
                float w = fmaxf(rs - 1.0f, 0.0f);          // (1-d)/d masked at d<1
                accS1[r] += w;
                accX[r]  += w * ox0;
                accY[r]  += w * oy0;
                accZ[r]  += w * oz0;
            }
            {   // tile 1
                float q  = fmaxf(d1[r], 1e-12f);
                float rs = __builtin_amdgcn_rsqf(q);
                rs = rs * (1.5f - 0.5f * q * rs * rs);
                float w = fmaxf(rs - 1.0f, 0.0f);
                accS1[r] += w;
                accX[r]  += w * ox1;
                accY[r]  += w * oy1;
                accZ[r]  += w * oz1;
            }
        }
    }

    // Reduce across the 16 column-lanes of each half (rows differ per half).
#pragma unroll
    for (int m = 1; m < 16; m <<= 1) {
#pragma unroll
        for (int r = 0; r < 8; ++r) {
            accS1[r] += __shfl_xor(accS1[r], m, 32);
            accX[r]  += __shfl_xor(accX[r],  m, 32);
            accY[r]  += __shfl_xor(accY[r],  m, 32);
            accZ[r]  += __shfl_xor(accZ[r],  m, 32);
        }
    }

    // One writer per (half, r): rows are padded to 64 so no bounds check needed.
#pragma unroll
    for (int r = 0; r < 8; ++r) {
        if (ln == r) {
            int row = mtile * 16 + r + 8 * hi;
            atomicAdd(&S1[row],         accS1[r]);
            atomicAdd(&So[row * 3 + 0], accX[r]);
            atomicAdd(&So[row * 3 + 1], accY[r]);
            atomicAdd(&So[row * 3 + 2], accZ[r]);
        }
    }
}

// ---------------------------------------------------------------------------
// Apply gradient step: smooth stencil + obstacle term, clamp endpoints,
// zero accumulators for the next iteration. Single block.
// ---------------------------------------------------------------------------
__global__ __launch_bounds__(512) void chomp_update(const float* __restrict__ start,
                                                    const float* __restrict__ goal,
                                                    float* __restrict__ ws,
                                                    float* __restrict__ out) {
    __shared__ float s[kT * kJ];
    const int i = threadIdx.x;
    if (i < kT * kJ) s[i] = ws[WS_TRAJ + i];
    __syncthreads();

    float nv = 0.0f;
    if (i < kT * kJ) {
        const int t = i / kJ, j = i % kJ;
        if (t == 0) {
            nv = start[j];
        } else if (t == kT - 1) {
            nv = goal[j];
        } else {
            const float invdt = (float)(kT - 1);          // 1/dt = 49
            auto acc = [&](int k) -> float {
                if (k < 0 || k > kT - 3) return 0.0f;
                float v1 = (s[(k + 1) * kJ + j] - s[k * kJ + j]) * invdt;
                float v2 = (s[(k + 2) * kJ + j] - s[(k + 1) * kJ + j]) * invdt;
                return (v2 - v1) * invdt;
            };
            // d/dtraj[t] of sum(acc^2)*dt = (2/dt) * (acc(t-2) - 2 acc(t-1) + acc(t))
            float g = (2.0f * invdt) * (acc(t - 2) - 2.0f * acc(t - 1) + acc(t));
            if (j < 3) {
                float S1v = ws[WS_S1 + t];
                float Sov = ws[WS_SO + t * 3 + j];
                g += -2.0f * kWObs * (S1v * s[i] - Sov);
            }
            nv = s[i] - kLR * g;
        }
    }
    __syncthreads();  // all reads of S1/So complete before zeroing

    if (i < kT * kJ) {
        ws[WS_TRAJ + i] = nv;
        if (out) out[i] = nv;
    }
    if (i < 256) ws[WS_S1 + i] = 0.0f;   // reset S1 + So for next iteration
}

// ---------------------------------------------------------------------------
extern "C" void kernel_launch(void* const* d_in, const int* in_sizes, int n_in,
                              void* d_out, int out_size, void* d_ws, size_t ws_size,
                              hipStream_t stream) {
    const float* start     = (const float*)d_in[0];
    const float* goal      = (const float*)d_in[1];
    const float* obstacles = (const float*)d_in[2];
    const int nObs    = in_sizes[2] / 3;
    const int nChunks = (nObs + 31) / 32;   // 32 obstacles (2 WMMA column-tiles) per trip

    float* ws  = (float*)d_ws;
    float* out = (float*)d_out;

    chomp_init<<<1, 512, 0, stream>>>(start, goal, ws);

    dim3 gridGrad(50, 4);  // 50 obstacle-strips x 4 trajectory row-tiles
    for (int it = 0; it < kIter; ++it) {
        chomp_grad<<<gridGrad, 256, 0, stream>>>(obstacles, ws + WS_TRAJ,
                                                 ws + WS_S1, ws + WS_SO,
                                                 nObs, nChunks);
        chomp_update<<<1, 512, 0, stream>>>(start, goal, ws,
                                            (it == kIter - 1) ? out : nullptr);
    }
}